// Half_Graph_80109730005370
// MI455X (gfx1250) — compile-verified
//
#include <hip/hip_runtime.h>
#include <hip/hip_bf16.h>
#include <math.h>

typedef float v2f __attribute__((ext_vector_type(2)));
typedef float v8f __attribute__((ext_vector_type(8)));

#define BB   8
#define CC   10
#define HH   256
#define WW   256
#define HWSZ (HH * WW)            // 65536
#define BCHW (BB * CC * HWSZ)     // 5242880

// Tile: one output row segment of 64 pixels; 4 waves x 16 pixels each.
#define TILE_W 64
#define HALO_W (TILE_W + 2)       // 66
#define CHSTRIDE (3 * HALO_W)     // 198 floats per LDS channel plane
#define NTHREADS 128

__global__ __launch_bounds__(NTHREADS)
void half_graph_fused_kernel(
    const float* __restrict__ f_nodes,   // (2,B,C,H,W)
    const float* __restrict__ h_nodes,   // (3,B,C,H,W)
    const float* __restrict__ p_nodes,   // (7,B,C,H,W)
    const float* __restrict__ rel_W,     // (4,10,20,3,3)
    const float* __restrict__ rel_gamma, // (4,10)
    const float* __restrict__ rel_beta,
    const float* __restrict__ rel_mean,
    const float* __restrict__ rel_var,
    const float* __restrict__ gg_W,      // (2,2,20)
    const float* __restrict__ gg_b,      // (2,2)
    const float* __restrict__ gc_W,      // (2,10,20)
    float* __restrict__ out)             // 3*BCHW : [h0][h1_new][h2_new]
{
    // ---- LDS ----
    __shared__ float sX[4 * 20 * CHSTRIDE];              // conv inputs, 4 rel x 20ch x 3rows x 66
    __shared__ __align__(16) float sW[4 * 16 * 180];     // weights, Cout padded to 16, K=180
    __shared__ float sXu[10 * TILE_W];                   // decomp0+comp_u
    __shared__ float sXl[10 * TILE_W];                   // decomp1+comp_l
    __shared__ float sBNi[4 * 16];                       // BN scale (pad ch>=10 -> 0)
    __shared__ float sBNs[4 * 16];                       // BN shift
    __shared__ float sGW[2 * 2 * 20];
    __shared__ float sGB[4];
    __shared__ float sCW[2 * 10 * 20];

    const int tid   = threadIdx.x;
    const int bx    = blockIdx.x;
    const int strip = bx & 3;            // W / TILE_W = 4
    const int y     = (bx >> 2) & (HH - 1);
    const int b     = bx >> 10;
    const int x0    = strip * TILE_W;

    const float* f1 = f_nodes + BCHW;        // f_nodes[1]
    const float* h1 = h_nodes + BCHW;        // h_nodes[1]
    const float* h2 = h_nodes + 2 * BCHW;    // h_nodes[2]

    // ---- Stage conv inputs (with halo + zero padding) ----
    // rel0: [h1, f1-h2]  rel1: [h2, f1-h1]  rel2: [h1, p1+p2+p3+p4]  rel3: [h2, p5+p6]
    for (int idx = tid; idx < 4 * 20 * CHSTRIDE; idx += NTHREADS) {
        const int col = idx % HALO_W;
        const int t1  = idx / HALO_W;
        const int row = t1 % 3;
        const int t2  = t1 / 3;
        const int ch  = t2 % 20;
        const int rel = t2 / 20;
        const int gy  = y - 1 + row;
        const int gx  = x0 - 1 + col;
        float v = 0.0f;
        if ((unsigned)gy < (unsigned)HH && (unsigned)gx < (unsigned)WW) {
            const int c   = (ch < 10) ? ch : (ch - 10);
            const int off = (b * CC + c) * HWSZ + (gy << 8) + gx;
            if (ch < 10) {
                v = (rel == 0 || rel == 2) ? h1[off] : h2[off];
            } else if (rel == 0) {
                v = f1[off] - h2[off];
            } else if (rel == 1) {
                v = f1[off] - h1[off];
            } else if (rel == 2) {
                v = p_nodes[1 * BCHW + off] + p_nodes[2 * BCHW + off]
                  + p_nodes[3 * BCHW + off] + p_nodes[4 * BCHW + off];
            } else {
                v = p_nodes[5 * BCHW + off] + p_nodes[6 * BCHW + off];
            }
        }
        sX[idx] = v;
    }

    // ---- Stage weights, K = tap*20 + cin, Cout padded to 16 with zeros ----
    for (int idx = tid; idx < 4 * 16 * 180; idx += NTHREADS) {
        const int k   = idx % 180;
        const int m   = (idx / 180) % 16;
        const int rel = idx / (180 * 16);
        float v = 0.0f;
        if (m < 10) {
            const int cin = k % 20;
            const int tap = k / 20;
            const int ky  = tap / 3;
            const int kx  = tap % 3;
            v = rel_W[(((rel * 10 + m) * 20 + cin) * 3 + ky) * 3 + kx];
        }
        sW[idx] = v;
    }

    // ---- Stage BN affine + GRU params ----
    if (tid < 64) {
        const int rel = tid >> 4;
        const int ch  = tid & 15;
        float inv = 0.0f, sh = 0.0f;
        if (ch < 10) {
            const int i = rel * 10 + ch;
            inv = rel_gamma[i] * rsqrtf(rel_var[i] + 1e-5f);
            sh  = rel_beta[i] - rel_mean[i] * inv;
        }
        sBNi[tid] = inv;
        sBNs[tid] = sh;
    }
    for (int idx = tid; idx < 80;  idx += NTHREADS) sGW[idx] = gg_W[idx];
    for (int idx = tid; idx < 400; idx += NTHREADS) sCW[idx] = gc_W[idx];
    if (tid < 4) sGB[tid] = gg_b[tid];

    __syncthreads();

    // ---- Implicit-GEMM conv via V_WMMA_F32_16X16X4_F32 ----
    // A: 16(Cout) x 4(K)  lane: M = lane&15, K = 2*(lane>>4) + {0,1}
    // B: 4(K) x 16(pix)   lane: N = lane&15, K = 2*(lane>>4) + {0,1}
    // D: lanes 0-15 N=lane M=r, lanes 16-31 N=lane-16 M=r+8
    const int lane = tid & 31;
    const int wv   = tid >> 5;
    const int n    = lane & 15;
    const int kh   = lane >> 4;
    const int pb   = wv * 16;

    v8f acc[4] = {v8f{}, v8f{}, v8f{}, v8f{}};

    // Lane-dependent bases; all per-fragment offsets below are compile-time
    // constants that fit the 16-bit DS immediate-offset field.
    const float* bB0 = sX + (kh * 2) * CHSTRIDE + pb + n;   // + ky*66 + kx per tap
    const float* bA0 = sW + n * 180 + kh * 2;               // + tap*20 per tap

    for (int tap = 0; tap < 9; ++tap) {
        const int ky = tap / 3;
        const int kx = tap % 3;
        const float* bB = bB0 + ky * HALO_W + kx;
        const float* bA = bA0 + tap * 20;
#pragma unroll
        for (int g = 0; g < 5; ++g) {
#pragma unroll
            for (int rel = 0; rel < 4; ++rel) {
                v2f A = *reinterpret_cast<const v2f*>(bA + rel * 16 * 180 + g * 4);
                v2f Bf;
                Bf.x = bB[(rel * 20 + g * 4)     * CHSTRIDE];
                Bf.y = bB[(rel * 20 + g * 4 + 1) * CHSTRIDE];
                acc[rel] = __builtin_amdgcn_wmma_f32_16x16x4_f32(
                    false, A, false, Bf, (short)0, acc[rel], false, false);
            }
        }
    }

    // ---- BN + ReLU, combine (decomp + comp), stash to LDS ----
#pragma unroll
    for (int r = 0; r < 8; ++r) {
        const int ch = r + 8 * kh;
        const float xu = fmaxf(acc[0][r] * sBNi[0 * 16 + ch] + sBNs[0 * 16 + ch], 0.0f)
                       + fmaxf(acc[2][r] * sBNi[2 * 16 + ch] + sBNs[2 * 16 + ch], 0.0f);
        const float xl = fmaxf(acc[1][r] * sBNi[1 * 16 + ch] + sBNs[1 * 16 + ch], 0.0f)
                       + fmaxf(acc[3][r] * sBNi[3 * 16 + ch] + sBNs[3 * 16 + ch], 0.0f);
        if (ch < 10) {
            sXu[ch * TILE_W + pb + n] = xu;
            sXl[ch * TILE_W + pb + n] = xl;
        }
    }

    __syncthreads();

    // ---- Pointwise ConvGRU: thread = (gru, pixel) ----
    {
        const int gru = tid >> 6;      // 0 -> h1, 1 -> h2
        const int px  = tid & 63;

        float xv[10], hv[10];
#pragma unroll
        for (int ch = 0; ch < 10; ++ch) {
            xv[ch] = (gru ? sXl : sXu)[ch * TILE_W + px];
            // h1 lives in sX rel0 node half, h2 in rel1 node half (middle row)
            hv[ch] = sX[(gru * 20 + ch) * CHSTRIDE + HALO_W + 1 + px];
        }

        float rs = sGB[gru * 2 + 0];
        float us = sGB[gru * 2 + 1];
#pragma unroll
        for (int k = 0; k < 20; ++k) {
            const float cmb = (k < 10) ? xv[k] : hv[k - 10];
            rs += sGW[(gru * 2 + 0) * 20 + k] * cmb;
            us += sGW[(gru * 2 + 1) * 20 + k] * cmb;
        }
        const float rg = 1.0f / (1.0f + __expf(-rs));
        const float ug = 1.0f / (1.0f + __expf(-us));

        const int obase = (1 + gru) * BCHW + (b * CC) * HWSZ + (y << 8) + x0 + px;
#pragma unroll
        for (int ch = 0; ch < 10; ++ch) {
            float s = 0.0f;
#pragma unroll
            for (int k = 0; k < 20; ++k) {
                const float cmb = (k < 10) ? xv[k] : rg * hv[k - 10];
                s += sCW[(gru * 10 + ch) * 20 + k] * cmb;
            }
            const float cn = tanhf(s);
            out[obase + ch * HWSZ] = (1.0f - ug) * hv[ch] + ug * cn;
        }
    }
}

extern "C" void kernel_launch(void* const* d_in, const int* in_sizes, int n_in,
                              void* d_out, int out_size, void* d_ws, size_t ws_size,
                              hipStream_t stream) {
    (void)in_sizes; (void)n_in; (void)out_size; (void)d_ws; (void)ws_size;
    const float* f_nodes   = (const float*)d_in[0];
    const float* h_nodes   = (const float*)d_in[1];
    const float* p_nodes   = (const float*)d_in[2];
    // d_in[3] = xh (unused)
    const float* rel_W     = (const float*)d_in[4];
    const float* rel_gamma = (const float*)d_in[5];
    const float* rel_beta  = (const float*)d_in[6];
    const float* rel_mean  = (const float*)d_in[7];
    const float* rel_var   = (const float*)d_in[8];
    const float* gg_W      = (const float*)d_in[9];
    const float* gg_b      = (const float*)d_in[10];
    const float* gc_W      = (const float*)d_in[11];
    float* out = (float*)d_out;

    // h0 passthrough: out[0 : BCHW] = h_nodes[0]
    hipMemcpyAsync(out, h_nodes, (size_t)BCHW * sizeof(float),
                   hipMemcpyDeviceToDevice, stream);

    const int blocks = BB * HH * (WW / TILE_W);   // 8192
    half_graph_fused_kernel<<<blocks, NTHREADS, 0, stream>>>(
        f_nodes, h_nodes, p_nodes, rel_W, rel_gamma, rel_beta, rel_mean, rel_var,
        gg_W, gg_b, gc_W, out);
}